// GatedElementBasedRNNLayer_61710090108940
// MI455X (gfx1250) — compile-verified
//
#include <hip/hip_runtime.h>
#include <hip/hip_bf16.h>

#define TPn   400
#define TQn   60
#define BATCH 16
#define IDIM  256
#define HDIM  256
#define TQPAD 64

typedef __attribute__((ext_vector_type(2))) float v2f;
typedef __attribute__((ext_vector_type(8))) float v8f;

enum { EP_NONE = 0, EP_GATE = 2 };

__device__ __forceinline__ float sigmoidf_(float x) { return 1.0f / (1.0f + __expf(-x)); }

// D = A(16x4) * B(4x16) + C   (fp32 WMMA, wave32)
__device__ __forceinline__ v8f wmma4(v2f a, v2f b, v8f c) {
  return __builtin_amdgcn_wmma_f32_16x16x4_f32(false, a, false, b, (short)0, c, false, false);
}

// ---------------------------------------------------------------------------
// Generic strided WMMA GEMM. One wave computes a 16x32 tile (two 16x16 C
// tiles sharing the A fragment -> 2-way WMMA ILP, half the A traffic).
//   C[m,n] = epilogue( sum_k A[m,k]*B[k,n] + bias[n] )
// AUNIT/BUNIT: compile-time "K-stride == 1" so the K-pairs lower to
// global_load_b64 with immediate offsets instead of scalar b32 + u64 adds.
// ---------------------------------------------------------------------------
template <bool AUNIT, bool BUNIT, int EPIL>
__global__ __launch_bounds__(32) void gemm_wmma_kernel(
    const float* __restrict__ A, long sAm, long sAk, long sAb,
    const float* __restrict__ Bm, long sBk, long sBn, long sBb,
    float* __restrict__ C, long sCm, long sCn, long sCb,
    const float* __restrict__ bias,
    const float* __restrict__ gateSrc, long sGm, long sGn, long sGb,
    int K)
{
  const int lane = threadIdx.x & 31;
  const int half = lane >> 4;     // 0: lanes 0-15, 1: lanes 16-31
  const int l    = lane & 15;
  const int m0 = blockIdx.x * 16;
  const int n0 = blockIdx.y * 32;      // two adjacent 16-wide N tiles
  const int bz = blockIdx.z;

  A  += (long)bz * sAb;
  Bm += (long)bz * sBb;
  C  += (long)bz * sCb;
  const float* G = gateSrc ? gateSrc + (long)bz * sGb : nullptr;

  const int nA = n0 + l;
  const int nB = n0 + 16 + l;
  v8f acc0, acc1;
  const float b0i = bias ? bias[nA] : 0.0f;
  const float b1i = bias ? bias[nB] : 0.0f;
#pragma unroll
  for (int g = 0; g < 8; ++g) { acc0[g] = b0i; acc1[g] = b1i; }

  const float* Arow  = A + (long)(m0 + l) * sAm;
  const float* Bcol0 = Bm + (long)nA * sBn;
  const float* Bcol1 = Bm + (long)nB * sBn;

  for (int k = 0; k < K; k += 4) {
    const int ka = k + 2 * half;   // per-lane K pair per ISA A/B layout
    v2f a, b0, b1;
    if (AUNIT) {
      a = *(const v2f*)(Arow + ka);               // global_load_b64
    } else {
      a[0] = Arow[(long)(ka + 0) * sAk];
      a[1] = Arow[(long)(ka + 1) * sAk];
    }
    if (BUNIT) {
      b0 = *(const v2f*)(Bcol0 + ka);             // global_load_b64
      b1 = *(const v2f*)(Bcol1 + ka);
    } else {
      b0[0] = Bcol0[(long)(ka + 0) * sBk];
      b0[1] = Bcol0[(long)(ka + 1) * sBk];
      b1[0] = Bcol1[(long)(ka + 0) * sBk];
      b1[1] = Bcol1[(long)(ka + 1) * sBk];
    }
    acc0 = wmma4(a, b0, acc0);
    acc1 = wmma4(a, b1, acc1);
  }

#pragma unroll
  for (int g = 0; g < 8; ++g) {
    const int m = m0 + g + 8 * half;
    float c0 = acc0[g], c1 = acc1[g];
    if (EPIL == EP_GATE) {
      c0 = G[(long)m * sGm + (long)nA * sGn] * sigmoidf_(c0);
      c1 = G[(long)m * sGm + (long)nB * sGn] * sigmoidf_(c1);
    }
    C[(long)m * sCm + (long)nA * sCn] = c0;
    C[(long)m * sCm + (long)nB * sCn] = c1;
  }
}

// ---------------------------------------------------------------------------
// Attention scores + masked softmax over q. One wave per (p,b) pair:
//   s_q = v . tanh(Qh[q,b,:] + Ph[p,b,:])
// float4 (b128) loads; writes a[(p*B+b)*64 + q], zero-padded q in [60,64).
// ---------------------------------------------------------------------------
__global__ __launch_bounds__(256) void attention_softmax_kernel(
    const float* __restrict__ Qh,    // (TQ,B,H)
    const float* __restrict__ Ph,    // (TP,B,H)
    const float* __restrict__ v,     // (H)
    const float* __restrict__ qmask, // (TQ,B)
    float* __restrict__ att)         // (TP,B,64)
{
  const int wave = threadIdx.x >> 5;
  const int lane = threadIdx.x & 31;
  const int pair = blockIdx.x * 8 + wave;
  __shared__ float sc[8][TQPAD];

  if (pair < TPn * BATCH) {
    const int p = pair >> 4, b = pair & 15;
    const float4* ph4 = (const float4*)(Ph + ((long)p * BATCH + b) * HDIM);
    const float4* v4  = (const float4*)v;
    const float4 pa = ph4[lane * 2], pb = ph4[lane * 2 + 1];
    const float4 va = v4[lane * 2],  vb = v4[lane * 2 + 1];
    for (int q = 0; q < TQn; ++q) {
      const float4* qh4 = (const float4*)(Qh + ((long)q * BATCH + b) * HDIM);
      const float4 qa = qh4[lane * 2], qb = qh4[lane * 2 + 1];
      float s = va.x * tanhf(qa.x + pa.x) + va.y * tanhf(qa.y + pa.y) +
                va.z * tanhf(qa.z + pa.z) + va.w * tanhf(qa.w + pa.w) +
                vb.x * tanhf(qb.x + pb.x) + vb.y * tanhf(qb.y + pb.y) +
                vb.z * tanhf(qb.z + pb.z) + vb.w * tanhf(qb.w + pb.w);
      for (int off = 16; off; off >>= 1) s += __shfl_xor(s, off, 32);
      if (lane == 0) {
        const float m = qmask[q * BATCH + b];
        sc[wave][q] = m * s + (1.0f - m) * (-1e30f);
      }
    }
  }
  __syncthreads();
  if (pair < TPn * BATCH) {
    float mx = -3.4e38f;
    for (int q = lane; q < TQn; q += 32) mx = fmaxf(mx, sc[wave][q]);
    for (int off = 16; off; off >>= 1) mx = fmaxf(mx, __shfl_xor(mx, off, 32));
    float sum = 0.0f;
    for (int q = lane; q < TQn; q += 32) sum += __expf(sc[wave][q] - mx);
    for (int off = 16; off; off >>= 1) sum += __shfl_xor(sum, off, 32);
    const float inv = 1.0f / sum;
    float* out = att + (long)pair * TQPAD;
    for (int q = lane; q < TQPAD; q += 32)
      out[q] = (q < TQn) ? __expf(sc[wave][q] - mx) * inv : 0.0f;
  }
}

// copy passage_repr into first half of the concat buffer x0[...,0:256]
__global__ void concat_copy_kernel(const float* __restrict__ src, float* __restrict__ x0) {
  const long i = (long)blockIdx.x * blockDim.x + threadIdx.x; // over TP*B*I
  if (i < (long)TPn * BATCH * IDIM) {
    const long row = i >> 8;
    const int  col = (int)(i & 255);
    x0[row * 512 + col] = src[i];
  }
}

// ---------------------------------------------------------------------------
// Bidirectional GRU scan for one layer. grid.x = direction (0 fwd, 1 bwd).
// 16 waves; per step each wave keeps 3 interleaved WMMA accumulators for its
// 3 N-tiles of gh = h @ Whh.T + bhh, sharing one ds_load_b64 A fragment per
// K-step (3-way matrix-pipe ILP on the serial recurrence chain). Then 512
// threads apply the gate math and update h in LDS. Reverse direction iterates
// t from TP-1 downward (equivalent to flip-scan-flip).
// ---------------------------------------------------------------------------
__global__ __launch_bounds__(512) void gru_scan_kernel(
    const float* __restrict__ xg,   // (2, TP, B, 3H)
    const float* __restrict__ Whh,  // (2, 3H, H) layer base
    const float* __restrict__ bhh,  // (2, 3H)
    float* __restrict__ y)          // (TP, B, 2H) concat output
{
  const int dir  = blockIdx.x;
  const int tid  = threadIdx.x;
  const int wave = tid >> 5;
  const int lane = tid & 31;
  const int half = lane >> 4;
  const int l    = lane & 15;

  __shared__ float h[BATCH][HDIM];        // 16 KB
  __shared__ float gh[BATCH][3 * HDIM];   // 48 KB

  const float* Wd  = Whh + (long)dir * 3 * HDIM * HDIM;
  const float* bd  = bhh + (long)dir * 3 * HDIM;
  const float* xgd = xg  + (long)dir * TPn * BATCH * 3 * HDIM;

  for (int i = tid; i < BATCH * HDIM; i += 512) h[i >> 8][i & 255] = 0.0f;
  __syncthreads();

  const int nb = wave * 48;           // this wave's 3 N-tiles: nb, nb+16, nb+32
  const float* w0 = Wd + (long)(nb + l) * HDIM;
  const float* w1 = w0 + 16 * HDIM;
  const float* w2 = w0 + 32 * HDIM;
  const float bv0 = bd[nb + l], bv1 = bd[nb + 16 + l], bv2 = bd[nb + 32 + l];

  for (int s = 0; s < TPn; ++s) {
    const int t = dir ? (TPn - 1 - s) : s;

    // Phase A: gh = h @ Whh.T + bhh  (M=16 batch, N=768, K=256)
    v8f acc0, acc1, acc2;
#pragma unroll
    for (int g = 0; g < 8; ++g) { acc0[g] = bv0; acc1[g] = bv1; acc2[g] = bv2; }
    for (int k = 0; k < HDIM; k += 4) {
      const int ka = k + 2 * half;
      const v2f a  = *(const v2f*)&h[l][ka];      // ds_load_b64, shared 3 ways
      const v2f b0 = *(const v2f*)(w0 + ka);      // global_load_b64 (L2-hot)
      const v2f b1 = *(const v2f*)(w1 + ka);
      const v2f b2 = *(const v2f*)(w2 + ka);
      acc0 = wmma4(a, b0, acc0);
      acc1 = wmma4(a, b1, acc1);
      acc2 = wmma4(a, b2, acc2);
    }
#pragma unroll
    for (int g = 0; g < 8; ++g) {
      const int m = g + 8 * half;
      gh[m][nb + l]      = acc0[g];
      gh[m][nb + 16 + l] = acc1[g];
      gh[m][nb + 32 + l] = acc2[g];
    }
    __syncthreads();

    // Phase B: gate math + state update, 512 threads x 8 elems = 16*256
    const float* xt = xgd + (long)t * BATCH * 3 * HDIM;
#pragma unroll
    for (int e = 0; e < 8; ++e) {
      const int idx = e * 512 + tid;
      const int b = idx >> 8, j = idx & 255;
      const float xr = xt[b * 768 + j];
      const float xz = xt[b * 768 + 256 + j];
      const float xn = xt[b * 768 + 512 + j];
      const float r  = sigmoidf_(xr + gh[b][j]);
      const float z  = sigmoidf_(xz + gh[b][256 + j]);
      const float nv = tanhf(xn + r * gh[b][512 + j]);
      const float hn = (1.0f - z) * nv + z * h[b][j];
      h[b][j] = hn;
      y[((long)t * BATCH + b) * 512 + dir * HDIM + j] = hn;
    }
    __syncthreads();
  }
}

// ---------------------------------------------------------------------------
extern "C" void kernel_launch(void* const* d_in, const int* in_sizes, int n_in,
                              void* d_out, int out_size, void* d_ws, size_t ws_size,
                              hipStream_t stream) {
  const float* passage  = (const float*)d_in[0];  // (400,16,256)
  const float* question = (const float*)d_in[1];  // (60,16,256)
  const float* qmask    = (const float*)d_in[3];  // (60,16)
  const float* WuQ      = (const float*)d_in[4];  // (256,256)
  const float* WuP      = (const float*)d_in[5];  // (256,256)
  const float* vvec     = (const float*)d_in[6];  // (256)
  const float* Wg       = (const float*)d_in[7];  // (512,512)
  const float* Wih      = (const float*)d_in[8];  // (3,2,768,512)
  const float* Whh      = (const float*)d_in[9];  // (3,2,768,256)
  const float* bih      = (const float*)d_in[10]; // (3,2,768)
  const float* bhh      = (const float*)d_in[11]; // (3,2,768)

  float* ws  = (float*)d_ws;
  float* Qh  = ws;                      ws += (long)TQn * BATCH * HDIM;
  float* Ph  = ws;                      ws += (long)TPn * BATCH * HDIM;
  float* att = ws;                      ws += (long)TPn * BATCH * TQPAD;
  float* x0  = ws;                      ws += (long)TPn * BATCH * 512;   // concat/gate src
  float* x1  = ws;                      ws += (long)TPn * BATCH * 512;
  float* x2  = ws;                      ws += (long)TPn * BATCH * 512;
  float* xg  = ws;                      ws += 2L * TPn * BATCH * 768;

  // Qh = question @ WuQ.T   (M=960, N=256, K=256)
  gemm_wmma_kernel<true, true, EP_NONE><<<dim3(TQn * BATCH / 16, HDIM / 32, 1), 32, 0, stream>>>(
      question, IDIM, 1, 0, WuQ, 1, IDIM, 0, Qh, HDIM, 1, 0,
      nullptr, nullptr, 0, 0, 0, IDIM);

  // Ph = passage @ WuP.T    (M=6400, N=256, K=256)
  gemm_wmma_kernel<true, true, EP_NONE><<<dim3(TPn * BATCH / 16, HDIM / 32, 1), 32, 0, stream>>>(
      passage, IDIM, 1, 0, WuP, 1, IDIM, 0, Ph, HDIM, 1, 0,
      nullptr, nullptr, 0, 0, 0, IDIM);

  // attention scores + softmax (one wave per (p,b))
  attention_softmax_kernel<<<TPn * BATCH / 8, 256, 0, stream>>>(Qh, Ph, vvec, qmask, att);

  // x0[...,0:256] = passage
  {
    const long n = (long)TPn * BATCH * IDIM;
    concat_copy_kernel<<<(unsigned)((n + 255) / 256), 256, 0, stream>>>(passage, x0);
  }

  // ct: per-batch GEMM  x0[...,256:512] = att(p,q) @ question(q,i)
  // M=400, N=256, K=64, batch=16; B K-stride is large -> BUNIT=false
  gemm_wmma_kernel<true, false, EP_NONE><<<dim3(TPn / 16, IDIM / 32, BATCH), 32, 0, stream>>>(
      att, (long)BATCH * TQPAD, 1, TQPAD,
      question, (long)BATCH * IDIM, 1, IDIM,
      x0 + IDIM, (long)BATCH * 512, 1, 512,
      nullptr, nullptr, 0, 0, 0, TQPAD);

  // gate: x1 = x0 * sigmoid(x0 @ Wg.T)  (M=6400, N=512, K=512)
  gemm_wmma_kernel<true, true, EP_GATE><<<dim3(TPn * BATCH / 16, 512 / 32, 1), 32, 0, stream>>>(
      x0, 512, 1, 0, Wg, 1, 512, 0, x1, 512, 1, 0,
      nullptr, x0, 512, 1, 0, 512);

  // 3 bidirectional GRU layers
  const float* xin = x1;
  for (int layer = 0; layer < 3; ++layer) {
    float* xout = (layer == 0) ? x2 : (layer == 1) ? x1 : (float*)d_out;
    for (int d = 0; d < 2; ++d) {
      const float* Wd = Wih + ((long)layer * 2 + d) * 768 * 512;
      const float* bd = bih + ((long)layer * 2 + d) * 768;
      // xg_d = xin @ Wih.T + bih  (M=6400, N=768, K=512)
      gemm_wmma_kernel<true, true, EP_NONE><<<dim3(TPn * BATCH / 16, 768 / 32, 1), 32, 0, stream>>>(
          xin, 512, 1, 0, Wd, 1, 512, 0,
          xg + (long)d * TPn * BATCH * 768, 768, 1, 0,
          bd, nullptr, 0, 0, 0, 512);
    }
    gru_scan_kernel<<<2, 512, 0, stream>>>(
        xg, Whh + (long)layer * 2 * 768 * 256, bhh + (long)layer * 2 * 768, xout);
    xin = xout;
  }
}